// dfsphModule_33552284516861
// MI455X (gfx1250) — compile-verified
//
#include <hip/hip_runtime.h>
#include <hip/hip_bf16.h>

// ---------------------------------------------------------------------------
// DFSPH scatter kernels for MI455X (gfx1250, wave32).
// Memory-bound edge scatter: stream edge data through LDS with CDNA5 async
// global->LDS loads (double buffered, non-temporal so the 168MB one-shot
// stream doesn't evict the ~15MB hot gather/atomic working set from L2),
// gather packed particle state from L2, scatter with global_atomic_add_f32.
// ---------------------------------------------------------------------------

#define NTHREADS 256
#define TILES 8
#define EDGES_PER_BLOCK (NTHREADS * TILES)

#define LDS_AS __attribute__((address_space(3)))
#define GLB_AS __attribute__((address_space(1)))

typedef int v2i __attribute__((vector_size(8)));

#define SPIKY_MAG (-76394.3726841f)   // -(30/pi)/h^3, h=0.05
#define DT_F      (0.002f)
#define DT2_F     (4.0e-6f)

#define CPOL_NT 1   // gfx12 CPol: TH[2:0]=1 -> non-temporal, scope=0

// --- CDNA5 async global->LDS copies (ASYNCcnt-tracked, NT-hinted) -----------
__device__ __forceinline__ void async_cp_b32(const void* g, void* l) {
#if __has_builtin(__builtin_amdgcn_global_load_async_to_lds_b32)
  __builtin_amdgcn_global_load_async_to_lds_b32(
      (GLB_AS int*)(unsigned long long)g,
      (LDS_AS int*)(unsigned)(unsigned long long)l, 0, CPOL_NT);
#else
  unsigned la = (unsigned)(unsigned long long)l;
  asm volatile("global_load_async_to_lds_b32 %0, %1, off"
               :: "v"(la), "v"(g) : "memory");
#endif
}

__device__ __forceinline__ void async_cp_b64(const void* g, void* l) {
#if __has_builtin(__builtin_amdgcn_global_load_async_to_lds_b64)
  __builtin_amdgcn_global_load_async_to_lds_b64(
      (GLB_AS v2i*)(unsigned long long)g,
      (LDS_AS v2i*)(unsigned)(unsigned long long)l, 0, CPOL_NT);
#else
  unsigned la = (unsigned)(unsigned long long)l;
  asm volatile("global_load_async_to_lds_b64 %0, %1, off"
               :: "v"(la), "v"(g) : "memory");
#endif
}

// asm waits double as compiler memory barriers
#define WAIT_ASYNC_4() asm volatile("s_wait_asynccnt 0x4" ::: "memory")
#define WAIT_ASYNC_0() asm volatile("s_wait_asynccnt 0x0" ::: "memory")
#define WAIT_DS_0()    asm volatile("s_wait_dscnt 0x0"    ::: "memory")

// ---------------------------------------------------------------------------
// Per-edge scatter (placed first in file so the disasm snippet shows it).
// Each block walks TILES tiles of 256 edges, double-buffering the streamed
// edge data (q, dir, i, j) through LDS with async loads; each lane stages
// exactly the bytes it later consumes, so no block barrier is required —
// only s_wait_asynccnt.
// ---------------------------------------------------------------------------
__global__ __launch_bounds__(NTHREADS) void dfsph_edge(
    const float* __restrict__ qArr, const float2* __restrict__ dirArr,
    const int* __restrict__ nbrI, const int* __restrict__ nbrJ,
    const float4* __restrict__ PA, const float4* __restrict__ PB,
    float* __restrict__ k1x, float* __restrict__ k1y, float* __restrict__ k2,
    float* __restrict__ out, int E) {
  __shared__ float  s_q[2][NTHREADS];
  __shared__ float2 s_d[2][NTHREADS];
  __shared__ int    s_i[2][NTHREADS];
  __shared__ int    s_j[2][NTHREADS];

  const int tid  = threadIdx.x;
  const int base = blockIdx.x * EDGES_PER_BLOCK;

  // stage tile 0
  {
    int e = base + tid;
    if (e < E) {
      async_cp_b32(qArr + e,   &s_q[0][tid]);
      async_cp_b64(dirArr + e, &s_d[0][tid]);
      async_cp_b32(nbrI + e,   &s_i[0][tid]);
      async_cp_b32(nbrJ + e,   &s_j[0][tid]);
    }
  }

#pragma unroll
  for (int t = 0; t < TILES; ++t) {
    const int buf = t & 1;
    // prefetch tile t+1 into the other buffer (if any edge of it is live)
    const bool pf = (t + 1 < TILES) && (base + (t + 1) * NTHREADS < E);
    if (pf) {
      int e1 = base + (t + 1) * NTHREADS + tid;
      WAIT_DS_0();  // prior reads of the buffer being overwritten are done
      if (e1 < E) {
        async_cp_b32(qArr + e1,   &s_q[buf ^ 1][tid]);
        async_cp_b64(dirArr + e1, &s_d[buf ^ 1][tid]);
        async_cp_b32(nbrI + e1,   &s_i[buf ^ 1][tid]);
        async_cp_b32(nbrJ + e1,   &s_j[buf ^ 1][tid]);
      }
      WAIT_ASYNC_4();  // tile t's 4 async ops retired (loads complete in order)
    } else {
      WAIT_ASYNC_0();
    }

    int e = base + t * NTHREADS + tid;
    if (e < E) {
      int   i  = s_i[buf][tid];
      int   j  = s_j[buf][tid];
      float q  = s_q[buf][tid];
      float2 dv = s_d[buf][tid];

      float om  = fminf(fmaxf(1.f - q, 0.f), 1.f);
      float mag = SPIKY_MAG * om * om;
      float gx = dv.x * mag, gy = dv.y * mag;
      float g2 = gx * gx + gy * gy;

      float4 paj = PA[j];                         // {aa, t2c, mass, p}_j
      float4 pbj = PB[j];                         // {vx, vy, ax, ay}_j
      float4 pbi = PB[i];
      float  p_i = ((const float*)PA)[4 * i + 3];

      // alpha pieces (workspace)
      unsafeAtomicAdd(&k1x[i], paj.x * gx);
      unsafeAtomicAdd(&k1y[i], paj.x * gy);
      unsafeAtomicAdd(&k2[i],  paj.y * g2);
      // source term -> out[:,1]
      float vg = (pbi.x - pbj.x) * gx + (pbi.y - pbj.y) * gy;
      unsafeAtomicAdd(&out[5 * i + 1], -DT_F * paj.x * vg);
      // pressure acceleration -> out[:,2:4]
      float co = -paj.z * (p_i + paj.w);
      unsafeAtomicAdd(&out[5 * i + 2], co * gx);
      unsafeAtomicAdd(&out[5 * i + 3], co * gy);
      // kernel sum -> out[:,4]
      float ag = (pbi.z - pbj.z) * gx + (pbi.w - pbj.w) * gy;
      unsafeAtomicAdd(&out[5 * i + 4], DT2_F * paj.x * ag);
    }
  }
}

// ---------------------------------------------------------------------------
// Pack per-particle gather state; zero accumulators.
// PA = {actualArea, actualArea^2/(area*rho), area*rho, p/(dens*rho)^2}
// PB = {velx, vely, accx, accy}
// ---------------------------------------------------------------------------
__global__ __launch_bounds__(NTHREADS) void dfsph_init(
    const float* __restrict__ area, const float* __restrict__ aarea,
    const float* __restrict__ rho, const float* __restrict__ dens,
    const float* __restrict__ p2, const float2* __restrict__ vel,
    const float2* __restrict__ acc,
    float4* __restrict__ PA, float4* __restrict__ PB,
    float* __restrict__ k1x, float* __restrict__ k1y, float* __restrict__ k2,
    float* __restrict__ out, int n) {
  int idx = blockIdx.x * NTHREADS + threadIdx.x;
  if (idx >= n) return;
  float a = area[idx], aa = aarea[idx], r = rho[idx];
  float d = dens[idx], pp = p2[idx];
  float dr = d * r;
  PA[idx] = make_float4(aa, aa * aa / (a * r), a * r, pp / (dr * dr));
  float2 v = vel[idx], ac = acc[idx];
  PB[idx] = make_float4(v.x, v.y, ac.x, ac.y);
  k1x[idx] = 0.f; k1y[idx] = 0.f; k2[idx] = 0.f;
  out[idx * 5 + 1] = 0.f; out[idx * 5 + 2] = 0.f;
  out[idx * 5 + 3] = 0.f; out[idx * 5 + 4] = 0.f;
}

// ---------------------------------------------------------------------------
// Finalize alpha with clip into out[:,0].
// ---------------------------------------------------------------------------
__global__ __launch_bounds__(NTHREADS) void dfsph_final(
    const float* __restrict__ area, const float* __restrict__ aarea,
    const float* __restrict__ rho,
    const float* __restrict__ k1x, const float* __restrict__ k1y,
    const float* __restrict__ k2, float* __restrict__ out, int n) {
  int idx = blockIdx.x * NTHREADS + threadIdx.x;
  if (idx >= n) return;
  float aa   = aarea[idx];
  float mass = area[idx] * rho[idx];
  float fac  = -DT2_F * aa;
  float x = k1x[idx], y = k1y[idx], s = k2[idx];
  float alpha = fac / mass * (x * x + y * y) + fac * s;
  alpha = fminf(fmaxf(alpha, -1.0f), -1e-7f);
  out[idx * 5] = alpha;
}

extern "C" void kernel_launch(void* const* d_in, const int* in_sizes, int n_in,
                              void* d_out, int out_size, void* d_ws, size_t ws_size,
                              hipStream_t stream) {
  const float*  area   = (const float*)d_in[0];
  const float*  aarea  = (const float*)d_in[1];
  const float*  rho    = (const float*)d_in[2];
  const float*  dens   = (const float*)d_in[3];
  const float*  p2     = (const float*)d_in[4];
  const float2* vel    = (const float2*)d_in[5];
  const float2* acc    = (const float2*)d_in[6];
  const float*  qArr   = (const float*)d_in[7];
  const float2* dirArr = (const float2*)d_in[8];
  const int*    nbr    = (const int*)d_in[9];   // int32: JAX x64 disabled

  const int N = in_sizes[0];
  const int E = in_sizes[7];
  const int* nbrI = nbr;
  const int* nbrJ = nbr + E;

  // workspace: k1x[N] k1y[N] k2[N] PA[4N] PB[4N]  (11N floats)
  float*  ws  = (float*)d_ws;
  float*  k1x = ws;
  float*  k1y = ws + (size_t)N;
  float*  k2  = ws + 2 * (size_t)N;
  float4* PA  = (float4*)(ws + 3 * (size_t)N);
  float4* PB  = PA + N;

  float* out = (float*)d_out;

  const int pBlocks = (N + NTHREADS - 1) / NTHREADS;
  const int eBlocks = (E + EDGES_PER_BLOCK - 1) / EDGES_PER_BLOCK;

  dfsph_init<<<pBlocks, NTHREADS, 0, stream>>>(
      area, aarea, rho, dens, p2, vel, acc, PA, PB, k1x, k1y, k2, out, N);
  dfsph_edge<<<eBlocks, NTHREADS, 0, stream>>>(
      qArr, dirArr, nbrI, nbrJ, PA, PB, k1x, k1y, k2, out, E);
  dfsph_final<<<pBlocks, NTHREADS, 0, stream>>>(
      area, aarea, rho, k1x, k1y, k2, out, N);
}